// STMetaLearner_old_54322746359862
// MI455X (gfx1250) — compile-verified
//
#include <hip/hip_runtime.h>
#include <hip/hip_bf16.h>

// ---------------- problem constants (match reference) ----------------
#define NTOT      16000      // B*NN nodes
#define EDGES     128000
#define NODE_DIM  32
#define EDGE_DIM  16
#define META_DIM  128
#define GAT_DIM   768
#define META_IN   800        // NODE_DIM + GAT_DIM
#define ZDIM      1616       // 2*META_IN + EDGE_DIM
#define NEG_SLOPE 0.2f

typedef float v2f __attribute__((ext_vector_type(2)));
typedef float v8f __attribute__((ext_vector_type(8)));

// D = A(16x4 f32) * B(4x16 f32) + C(16x16 f32), wave32, exact f32 semantics.
__device__ __forceinline__ v8f wmma4(v2f a, v2f b, v8f c) {
  return __builtin_amdgcn_wmma_f32_16x16x4_f32(
      /*neg_a=*/false, a, /*neg_b=*/false, b,
      /*c_mod=*/(short)0, c, /*reuse_a=*/false, /*reuse_b=*/false);
}

__device__ __forceinline__ void atomic_fadd(float* p, float v) {
  unsafeAtomicAdd(p, v);   // native global_atomic_add_f32 (no CAS loop)
}

// ---------------- init: x[:,0:768]=b_gat, amax=-inf, denom=0 ----------------
__global__ void k_init(float* __restrict__ x, const float* __restrict__ b_gat,
                       float* __restrict__ amax, float* __restrict__ denom) {
  int idx = blockIdx.x * blockDim.x + threadIdx.x;
  if (idx >= NTOT * GAT_DIM) return;
  int n = idx / GAT_DIM, d = idx - n * GAT_DIM;
  x[(size_t)n * META_IN + d] = b_gat[d];
  if (d == 0) {
    amax[n]  = __int_as_float(0xFF800000u);  // -inf
    denom[n] = 0.f;
  }
}

// ---------------- node learner: x[:,768:800] = nf @ W_node + b ----------------
__global__ void k_node_learner(const float* __restrict__ nfeat,
                               const float* __restrict__ W,
                               const float* __restrict__ b,
                               float* __restrict__ x) {
  int idx = blockIdx.x * blockDim.x + threadIdx.x;   // NTOT*32
  if (idx >= NTOT * NODE_DIM) return;
  int n = idx >> 5, o = idx & 31;
  const float* r = nfeat + (size_t)n * NODE_DIM;
  float acc = b[o];
#pragma unroll
  for (int k = 0; k < NODE_DIM; ++k) acc += r[k] * W[k * NODE_DIM + o];
  x[(size_t)n * META_IN + GAT_DIM + o] = acc;
}

// ---------------- edge learner: ea = edge_attr @ W_edge + b ----------------
__global__ void k_edge_learner(const float* __restrict__ eattr,
                               const float* __restrict__ W,
                               const float* __restrict__ b,
                               float* __restrict__ ea) {
  int idx = blockIdx.x * blockDim.x + threadIdx.x;   // EDGES*16
  if (idx >= EDGES * EDGE_DIM) return;
  int e = idx >> 4, o = idx & 15;
  const float* r = eattr + (size_t)e * EDGE_DIM;
  float acc = b[o];
#pragma unroll
  for (int k = 0; k < EDGE_DIM; ++k) acc += r[k] * W[k * EDGE_DIM + o];
  ea[(size_t)e * EDGE_DIM + o] = acc;
}

// ---------------- h = mf @ W_gat : WMMA f32, 1 wave = 16x128 tile ----------------
__global__ void k_gat_hgemm(const float* __restrict__ mf,
                            const float* __restrict__ Wg,
                            float* __restrict__ h) {
  int lane = threadIdx.x;
  int m = lane & 15, hi = lane >> 4;
  int row = blockIdx.x * 16 + m;
  int colBase = blockIdx.y * 128;
  const float* arow = mf + (size_t)row * GAT_DIM;

  v8f acc[8];
#pragma unroll
  for (int t = 0; t < 8; ++t)
#pragma unroll
    for (int v = 0; v < 8; ++v) acc[t][v] = 0.f;

  for (int k = 0; k < GAT_DIM; k += 4) {
    int ka = k + 2 * hi;
    v2f a = *(const v2f*)(arow + ka);
#pragma unroll
    for (int t = 0; t < 8; ++t) {
      int c = colBase + t * 16 + m;
      v2f bf;
      bf.x = Wg[(size_t)ka * GAT_DIM + c];
      bf.y = Wg[(size_t)(ka + 1) * GAT_DIM + c];
      acc[t] = wmma4(a, bf, acc[t]);
    }
  }
#pragma unroll
  for (int t = 0; t < 8; ++t)
#pragma unroll
    for (int v = 0; v < 8; ++v)
      h[(size_t)(blockIdx.x * 16 + v + 8 * hi) * GAT_DIM + colBase + t * 16 + m] = acc[t][v];
}

// ---------------- a_src/a_dst: wave-per-node dual dot ----------------
__global__ void k_att(const float* __restrict__ h,
                      const float* __restrict__ att_s, const float* __restrict__ att_d,
                      float* __restrict__ a_s, float* __restrict__ a_d) {
  int node = (blockIdx.x * blockDim.x + threadIdx.x) >> 5;
  int lane = threadIdx.x & 31;
  if (node >= NTOT) return;
  const float* r = h + (size_t)node * GAT_DIM;
  float s = 0.f, d = 0.f;
  for (int k = lane; k < GAT_DIM; k += 32) {
    float v = r[k];
    s += v * att_s[k];
    d += v * att_d[k];
  }
#pragma unroll
  for (int off = 16; off > 0; off >>= 1) {
    s += __shfl_xor(s, off, 32);
    d += __shfl_xor(d, off, 32);
  }
  if (lane == 0) { a_s[node] = s; a_d[node] = d; }
}

// ---------------- pass 1: alpha = lrelu(as[src]+ad[dst]); segment max ----------------
__global__ void k_alpha_max(const int* __restrict__ ei,
                            const float* __restrict__ a_s, const float* __restrict__ a_d,
                            float* __restrict__ alpha, float* __restrict__ amax) {
  int i = blockIdx.x * blockDim.x + threadIdx.x;
  if (i >= EDGES + NTOT) return;
  int s, d;
  if (i < EDGES) { s = ei[i]; d = ei[EDGES + i]; } else { s = d = i - EDGES; }
  float a = a_s[s] + a_d[d];
  a = a > 0.f ? a : NEG_SLOPE * a;
  alpha[i] = a;
  // float segment-max via ordered-int trick (amax initialized to -inf)
  if (a >= 0.f) atomicMax((int*)(amax + d), __float_as_int(a));
  else          atomicMin((unsigned int*)(amax + d), __float_as_uint(a));
}

// ---------------- pass 2: ex = exp(alpha - amax[dst]); denom += ex ----------------
__global__ void k_alpha_exp(const int* __restrict__ ei,
                            float* __restrict__ alpha,
                            const float* __restrict__ amax, float* __restrict__ denom) {
  int i = blockIdx.x * blockDim.x + threadIdx.x;
  if (i >= EDGES + NTOT) return;
  int d = (i < EDGES) ? ei[EDGES + i] : (i - EDGES);
  float e = __expf(alpha[i] - amax[d]);
  alpha[i] = e;
  atomic_fadd(denom + d, e);
}

// ---------------- pass 3: x[dst,0:768] += h[src] * (ex/denom[dst]) ----------------
__global__ void k_gat_scatter(const int* __restrict__ ei,
                              const float* __restrict__ alpha, const float* __restrict__ denom,
                              const float* __restrict__ h, float* __restrict__ x) {
  int item = blockIdx.x * (blockDim.x >> 5) + (threadIdx.x >> 5);
  int lane = threadIdx.x & 31;
  if (item >= EDGES + NTOT) return;
  int s, d;
  if (item < EDGES) { s = ei[item]; d = ei[EDGES + item]; } else { s = d = item - EDGES; }
  float coef = alpha[item] / denom[d];
  const float* hr = h + (size_t)s * GAT_DIM;
  float* xr = x + (size_t)d * META_IN;
  for (int c = lane; c < GAT_DIM; c += 32) atomic_fadd(xr + c, hr[c] * coef);
}

// ---------------- out = x @ W_self + b_self : WMMA, 1 wave = 16x128 ----------------
__global__ void k_self_gemm(const float* __restrict__ x,
                            const float* __restrict__ Ws, const float* __restrict__ bs,
                            float* __restrict__ out) {
  int lane = threadIdx.x;
  int m = lane & 15, hi = lane >> 4;
  int row = blockIdx.x * 16 + m;
  const float* arow = x + (size_t)row * META_IN;

  v8f acc[8];
#pragma unroll
  for (int t = 0; t < 8; ++t) {
    float bias = bs[t * 16 + m];
#pragma unroll
    for (int v = 0; v < 8; ++v) acc[t][v] = bias;
  }
  for (int k = 0; k < META_IN; k += 4) {
    int ka = k + 2 * hi;
    v2f a = *(const v2f*)(arow + ka);
#pragma unroll
    for (int t = 0; t < 8; ++t) {
      int c = t * 16 + m;
      v2f bf;
      bf.x = Ws[(size_t)ka * META_DIM + c];
      bf.y = Ws[(size_t)(ka + 1) * META_DIM + c];
      acc[t] = wmma4(a, bf, acc[t]);
    }
  }
#pragma unroll
  for (int t = 0; t < 8; ++t)
#pragma unroll
    for (int v = 0; v < 8; ++v)
      out[(size_t)(blockIdx.x * 16 + v + 8 * hi) * META_DIM + t * 16 + m] = acc[t][v];
}

// ---- msg = [x[dst]|x[src]|ea] @ W_nbr + b_nbr, scatter-add to out[dst] ----
__global__ void k_msg_gemm(const float* __restrict__ x, const float* __restrict__ ea,
                           const int* __restrict__ ei,
                           const float* __restrict__ Wn, const float* __restrict__ bn,
                           float* __restrict__ out) {
  int lane = threadIdx.x;
  int m = lane & 15, hi = lane >> 4;
  int e0 = blockIdx.x * 16;

  int eA = e0 + m;                       // row this lane gathers for A
  int srcE = ei[eA], dstE = ei[EDGES + eA];
  const float* rowDst = x + (size_t)dstE * META_IN;
  const float* rowSrc = x + (size_t)srcE * META_IN;
  const float* rowEa  = ea + (size_t)eA * EDGE_DIM;

  v8f acc[8];
#pragma unroll
  for (int t = 0; t < 8; ++t) {
    float bias = bn[t * 16 + m];
#pragma unroll
    for (int v = 0; v < 8; ++v) acc[t][v] = bias;
  }

  for (int k = 0; k < ZDIM; k += 4) {
    int ka = k + 2 * hi;
    v2f a;
    if (ka < META_IN)            a = *(const v2f*)(rowDst + ka);
    else if (ka < 2 * META_IN)   a = *(const v2f*)(rowSrc + (ka - META_IN));
    else                         a = *(const v2f*)(rowEa + (ka - 2 * META_IN));
#pragma unroll
    for (int t = 0; t < 8; ++t) {
      int c = t * 16 + m;
      v2f bf;
      bf.x = Wn[(size_t)ka * META_DIM + c];
      bf.y = Wn[(size_t)(ka + 1) * META_DIM + c];
      acc[t] = wmma4(a, bf, acc[t]);
    }
  }

  int dr[8];
#pragma unroll
  for (int v = 0; v < 8; ++v) dr[v] = ei[EDGES + e0 + v + 8 * hi];
#pragma unroll
  for (int t = 0; t < 8; ++t)
#pragma unroll
    for (int v = 0; v < 8; ++v)
      atomic_fadd(out + (size_t)dr[v] * META_DIM + t * 16 + m, acc[t][v]);
}

// ---------------- host-side launcher ----------------
extern "C" void kernel_launch(void* const* d_in, const int* in_sizes, int n_in,
                              void* d_out, int out_size, void* d_ws, size_t ws_size,
                              hipStream_t stream) {
  const float* node_feature = (const float*)d_in[0];
  const float* edge_attr    = (const float*)d_in[1];
  const float* message_feat = (const float*)d_in[2];
  const int*   edge_index   = (const int*)d_in[3];
  const float* W_node = (const float*)d_in[4];
  const float* b_node = (const float*)d_in[5];
  const float* W_edge = (const float*)d_in[6];
  const float* b_edge = (const float*)d_in[7];
  const float* W_gat  = (const float*)d_in[8];
  const float* att_src = (const float*)d_in[9];
  const float* att_dst = (const float*)d_in[10];
  const float* b_gat  = (const float*)d_in[11];
  const float* W_nbr  = (const float*)d_in[12];
  const float* b_nbr  = (const float*)d_in[13];
  const float* W_self = (const float*)d_in[14];
  const float* b_self = (const float*)d_in[15];
  float* out = (float*)d_out;

  // workspace layout (floats)
  float* h     = (float*)d_ws;                     // NTOT*768
  float* x     = h + (size_t)NTOT * GAT_DIM;       // NTOT*800
  float* ea    = x + (size_t)NTOT * META_IN;       // EDGES*16
  float* a_s   = ea + (size_t)EDGES * EDGE_DIM;    // NTOT
  float* a_d   = a_s + NTOT;                       // NTOT
  float* amax  = a_d + NTOT;                       // NTOT
  float* denom = amax + NTOT;                      // NTOT
  float* alpha = denom + NTOT;                     // EDGES+NTOT

  k_init<<<(NTOT * GAT_DIM + 255) / 256, 256, 0, stream>>>(x, b_gat, amax, denom);
  k_node_learner<<<(NTOT * NODE_DIM + 255) / 256, 256, 0, stream>>>(node_feature, W_node, b_node, x);
  k_edge_learner<<<(EDGES * EDGE_DIM + 255) / 256, 256, 0, stream>>>(edge_attr, W_edge, b_edge, ea);

  k_gat_hgemm<<<dim3(NTOT / 16, GAT_DIM / 128), 32, 0, stream>>>(message_feat, W_gat, h);
  k_att<<<(NTOT * 32 + 255) / 256, 256, 0, stream>>>(h, att_src, att_dst, a_s, a_d);

  int nitems = EDGES + NTOT;
  k_alpha_max<<<(nitems + 255) / 256, 256, 0, stream>>>(edge_index, a_s, a_d, alpha, amax);
  k_alpha_exp<<<(nitems + 255) / 256, 256, 0, stream>>>(edge_index, alpha, amax, denom);
  k_gat_scatter<<<(nitems * 32 + 255) / 256, 256, 0, stream>>>(edge_index, alpha, denom, h, x);

  k_self_gemm<<<dim3(NTOT / 16, 1), 32, 0, stream>>>(x, W_self, b_self, out);
  k_msg_gemm<<<dim3(EDGES / 16, 1), 32, 0, stream>>>(x, ea, edge_index, W_nbr, b_nbr, out);
}